// Conv_and_Sum_77738908058130
// MI455X (gfx1250) — compile-verified
//
#include <hip/hip_runtime.h>
#include <hip/hip_bf16.h>
#include <math.h>

typedef float v2f __attribute__((ext_vector_type(2)));
typedef float v4f __attribute__((ext_vector_type(4)));
typedef float v8f __attribute__((ext_vector_type(8)));

// ---------------------------------------------------------------------------
// Kernel 1: s[b,c] = sum_l x[b,c,l].  One 256-thread block per (b,c) row.
// x row is 2048 contiguous floats = 512 float4; each thread takes 2 float4.
// Nontemporal b128 loads: the 256 MB stream is single-use.
// ---------------------------------------------------------------------------
__global__ void __launch_bounds__(256)
row_sum_kernel(const float* __restrict__ x, float* __restrict__ s) {
    const int row = blockIdx.x;                       // 0 .. 32*1024-1
    const v4f* xr = (const v4f*)(x + (size_t)row * 2048);
    const int t = threadIdx.x;

    v4f a = __builtin_nontemporal_load(&xr[t]);
    v4f b = __builtin_nontemporal_load(&xr[t + 256]);
    float v = (a[0] + a[1]) + (a[2] + a[3]) + (b[0] + b[1]) + (b[2] + b[3]);

    // wave32 tree reduction
    #pragma unroll
    for (int off = 16; off > 0; off >>= 1)
        v += __shfl_down(v, off, 32);

    __shared__ float lds[8];
    const int wave = t >> 5;
    const int lane = t & 31;
    if (lane == 0) lds[wave] = v;
    __syncthreads();

    if (t == 0) {
        float acc = lds[0];
        #pragma unroll
        for (int i = 1; i < 8; ++i) acc += lds[i];
        s[row] = acc;
    }
}

// ---------------------------------------------------------------------------
// Kernel 2: y[b,o] = sum_c s[b,c] * W[o,c] + L*bias[o]; out = mish(y).
// One wave per 16x16 tile of the 32x1024 output. K accumulated 4 at a time
// with V_WMMA_F32_16X16X4_F32.
//
// A (16x4 f32) layout:  lanes 0-15: M=lane,    v0=K0, v1=K1
//                       lanes16-31: M=lane-16, v0=K2, v1=K3
// B (4x16 f32) layout:  lanes 0-15: N=lane,    v0=K0, v1=K1
//                       lanes16-31: N=lane-16, v0=K2, v1=K3
// C/D (16x16 f32):      VGPR r: lanes 0-15 -> M=r, lanes 16-31 -> M=r+8
// ---------------------------------------------------------------------------
__device__ __forceinline__ float mish_f(float y) {
    // softplus with overflow guard; tanh via libdevice
    float sp = (y > 20.0f) ? y : log1pf(__expf(y));
    return y * tanhf(sp);
}

__global__ void __launch_bounds__(32)
gemm_bias_mish_kernel(const float* __restrict__ s,     // [32,1024]
                      const float* __restrict__ W,     // [1024,1024] row-major [o][c]
                      const float* __restrict__ bias,  // [1024]
                      float* __restrict__ out)         // [32,1024]
{
    const int nt   = blockIdx.x;     // 0..63 : output-column tile (o)
    const int mb   = blockIdx.y;     // 0..1  : output-row tile (b)
    const int lane = threadIdx.x;    // 0..31
    const int half = lane >> 4;      // 0 or 1
    const int lrow = lane & 15;

    const int m0 = mb * 16;
    const int n0 = nt * 16;

    const float* srow = s + (size_t)(m0 + lrow) * 1024;  // A row: b = m0+lrow
    const float* wrow = W + (size_t)(n0 + lrow) * 1024;  // B col: o = n0+lrow

    v8f c = {};

    #pragma unroll 4
    for (int k0 = 0; k0 < 1024; k0 += 4) {
        const int k = k0 + half * 2;                 // this lane's K pair
        const v2f av = *(const v2f*)(srow + k);
        const v2f bv = *(const v2f*)(wrow + k);
        // 8 args: (neg_a, A, neg_b, B, c_mod, C, reuse_a, reuse_b)
        c = __builtin_amdgcn_wmma_f32_16x16x4_f32(
                false, av, false, bv, (short)0, c, false, false);
    }

    const int col = n0 + lrow;                       // o index for this lane
    const float bb = 2048.0f * bias[col];

    #pragma unroll
    for (int r = 0; r < 8; ++r) {
        const int row = m0 + r + 8 * half;           // b index
        const float y = c[r] + bb;
        out[(size_t)row * 1024 + col] = mish_f(y);
    }
}

// ---------------------------------------------------------------------------
// Launch
// ---------------------------------------------------------------------------
extern "C" void kernel_launch(void* const* d_in, const int* in_sizes, int n_in,
                              void* d_out, int out_size, void* d_ws, size_t ws_size,
                              hipStream_t stream) {
    const float* x    = (const float*)d_in[0];   // [32,1024,2048]
    const float* W    = (const float*)d_in[1];   // [1024,1024]
    const float* bias = (const float*)d_in[2];   // [1024]
    float* out        = (float*)d_out;           // [32,1024]
    float* s          = (float*)d_ws;            // 32*1024 floats = 128 KB scratch

    // Phase 1: bandwidth-bound row reduction (256 MB stream -> ~11 us at 23.3 TB/s)
    row_sum_kernel<<<32 * 1024, 256, 0, stream>>>(x, s);

    // Phase 2: tiny WMMA GEMM + bias + mish (128 waves)
    dim3 grid2(64, 2);
    gemm_bias_mish_kernel<<<grid2, 32, 0, stream>>>(s, W, bias, out);
}